// TSPEnergyFunction_61306363183464
// MI455X (gfx1250) — compile-verified
//
#include <hip/hip_runtime.h>
#include <math.h>

typedef __attribute__((ext_vector_type(2))) float v2f;
typedef __attribute__((ext_vector_type(4))) float v4f;
typedef __attribute__((ext_vector_type(8))) float v8f;

#define N_ROWS 8192
#define M_ROWS 8192
#define D_DIM  256
#define C_DIM  128
#define H_DIM  128
#define LAMBDA_IB 0.01f
#define EPS_N 1e-8f

// ---------------- Phase 1: row L2-normalize + swizzle into WMMA fragment layout ---------
// Output layout: for 16-row tile t, k-group g (k=8g), half hi, lane-row m:
//   slot(t,g,hi,m) = 4 floats = row (t*16+m), columns {2*(4g+hi), 2*(4g+hi)+1,
//                                                      2*(4g+2+hi), 2*(4g+2+hi)+1}
// flat float index = ((t*64 + g*2 + hi)*16 + m)*4 + j*2 + lo
// so that one global_load_b128 per lane (lanes 0-15 = hi 0, 16-31 = hi 1) is a fully
// contiguous 512B fetch feeding TWO 16x16x4 WMMA k-steps.
__global__ void rownorm_swizzle_kernel(const float* __restrict__ x, float* __restrict__ xsw) {
  __shared__ float red[256];
  const int row = blockIdx.x;
  const int c = threadIdx.x;              // blockDim = 256 = D_DIM
  const float v = x[(size_t)row * D_DIM + c];
  red[c] = v * v;
  __syncthreads();
  for (int s = 128; s > 0; s >>= 1) {
    if (c < s) red[c] += red[c + s];
    __syncthreads();
  }
  const float norm = fmaxf(sqrtf(red[0]), EPS_N);

  const int t  = row >> 4;
  const int m  = row & 15;
  const int p  = c >> 1;
  const int lo = c & 1;
  const int g  = p >> 2;
  const int hi = p & 1;
  const int j  = (p >> 1) & 1;
  const size_t idx = ((size_t)(t * 64 + g * 2 + hi) * 16 + m) * 4 + j * 2 + lo;
  xsw[idx] = v / norm;
}

// ---------------- Phase 2: compressor xc = x @ Wc + bc ----------------
__global__ void compress_kernel(const float* __restrict__ x, const float* __restrict__ Wc,
                                const float* __restrict__ bc, float* __restrict__ xc) {
  __shared__ float zrow[D_DIM];
  const int row = blockIdx.x;
  const int t = threadIdx.x;              // blockDim = 128 = C_DIM
  zrow[t]       = x[(size_t)row * D_DIM + t];
  zrow[t + 128] = x[(size_t)row * D_DIM + t + 128];
  __syncthreads();
  float acc = bc[t];
  #pragma unroll 8
  for (int k = 0; k < D_DIM; ++k) acc += zrow[k] * Wc[k * C_DIM + t];
  xc[(size_t)row * C_DIM + t] = acc;
}

// ---------------- Phase 3: fused critic MLP, one row per block ----------------
__global__ void critic_kernel(const float* __restrict__ xc1, const float* __restrict__ xc2,
                              const float* __restrict__ W1, const float* __restrict__ b1,
                              const float* __restrict__ W2, const float* __restrict__ b2,
                              const float* __restrict__ W3, const float* __restrict__ b3,
                              const int* __restrict__ perm, int use_perm,
                              float* __restrict__ t_out) {
  __shared__ float z[2 * C_DIM];
  __shared__ float h[H_DIM];
  const int row = blockIdx.x;
  const int t = threadIdx.x;              // blockDim = 128 = H_DIM
  const int j = use_perm ? perm[row] : row;
  z[t]         = xc1[(size_t)row * C_DIM + t];
  z[C_DIM + t] = xc2[(size_t)j * C_DIM + t];
  __syncthreads();
  float acc = b1[t];
  #pragma unroll 8
  for (int k = 0; k < 2 * C_DIM; ++k) acc += z[k] * W1[k * H_DIM + t];
  h[t] = fmaxf(acc, 0.0f);
  __syncthreads();
  acc = b2[t];
  #pragma unroll 8
  for (int k = 0; k < H_DIM; ++k) acc += h[k] * W2[k * H_DIM + t];
  const float h2 = fmaxf(acc, 0.0f);
  __syncthreads();
  z[t] = h2 * W3[t];                      // reuse z for reduction
  __syncthreads();
  for (int s = 64; s > 0; s >>= 1) {
    if (t < s) z[t] += z[t + s];
    __syncthreads();
  }
  if (t == 0) t_out[row] = z[0] + b3[0];
}

// ---------------- Phase 4: mi = mean(t_xy) - (mean(exp(t_sh)) + log 2) ----------------
__global__ void mi_reduce_kernel(const float* __restrict__ t_xy, const float* __restrict__ t_sh,
                                 float* __restrict__ mi) {
  __shared__ float s1[256];
  __shared__ float s2[256];
  const int t = threadIdx.x;              // blockDim = 256, single block
  float a = 0.0f, b = 0.0f;
  for (int i = t; i < N_ROWS; i += 256) {
    a += t_xy[i];
    b += expf(t_sh[i]);
  }
  s1[t] = a; s2[t] = b;
  __syncthreads();
  for (int s = 128; s > 0; s >>= 1) {
    if (t < s) { s1[t] += s1[t + s]; s2[t] += s2[t + s]; }
    __syncthreads();
  }
  if (t == 0)
    mi[0] = s1[0] / (float)N_ROWS - (s2[0] / (float)N_ROWS + 0.6931471805599453f);
}

// ---------------- Phase 5: distances via V_WMMA_F32_16X16X4_F32 ----------------
// Block = 256 threads = 8 waves laid out 4(M) x 2(N); wave tile = 64x32 (4x2 accum tiles).
// Block tile = 256 x 64. Inputs read from the swizzled fragment layout: each
// global_load_b128 is a fully contiguous 512B fetch providing TWO k-steps of fragments.
// Per k-group: 6 b128 loads feed 16 WMMAs.
__global__ void __launch_bounds__(256)
dist_wmma_kernel(const float* __restrict__ x1sw, const float* __restrict__ x2sw,
                 const float* __restrict__ mi_p, float* __restrict__ out) {
  const int lane   = threadIdx.x & 31;
  const int wave   = threadIdx.x >> 5;   // 0..7
  const int waveM  = wave & 3;           // 0..3
  const int waveN  = wave >> 2;          // 0..1
  const int laneM  = lane & 15;          // row-in-tile for A, col-in-tile for B
  const int laneHi = lane >> 4;          // K-pair half (ISA Sec 7.12.2 f32 A/B layout)

  const int rowWave = blockIdx.y * 256 + waveM * 64;
  const int colWave = blockIdx.x * 64  + waveN * 32;

  v8f acc[4][2];
  #pragma unroll
  for (int mt = 0; mt < 4; ++mt)
    #pragma unroll
    for (int nt = 0; nt < 2; ++nt)
      acc[mt][nt] = (v8f){0.f, 0.f, 0.f, 0.f, 0.f, 0.f, 0.f, 0.f};

  // slot base: ((tile*64 + g*2 + hi)*16 + m)*4 floats ; per-g advance = 128 floats
  const float* aBase[4];
  const float* bBase[2];
  #pragma unroll
  for (int mt = 0; mt < 4; ++mt)
    aBase[mt] = x1sw + ((size_t)(((rowWave >> 4) + mt) * 64 + laneHi) * 16 + laneM) * 4;
  #pragma unroll
  for (int nt = 0; nt < 2; ++nt)
    bBase[nt] = x2sw + ((size_t)(((colWave >> 4) + nt) * 64 + laneHi) * 16 + laneM) * 4;

  #pragma unroll 2
  for (int g = 0; g < D_DIM / 8; ++g) {  // k = 8g, two 16x16x4 steps per group
    v4f af[4], bf[2];
    #pragma unroll
    for (int mt = 0; mt < 4; ++mt) af[mt] = *(const v4f*)(aBase[mt] + g * 128);
    #pragma unroll
    for (int nt = 0; nt < 2; ++nt) bf[nt] = *(const v4f*)(bBase[nt] + g * 128);
    #pragma unroll
    for (int mt = 0; mt < 4; ++mt)
      #pragma unroll
      for (int nt = 0; nt < 2; ++nt) {
        v2f a0 = {af[mt].x, af[mt].y};
        v2f b0 = {bf[nt].x, bf[nt].y};
        acc[mt][nt] = __builtin_amdgcn_wmma_f32_16x16x4_f32(
            false, a0, false, b0, (short)0, acc[mt][nt], false, false);
      }
    #pragma unroll
    for (int mt = 0; mt < 4; ++mt)
      #pragma unroll
      for (int nt = 0; nt < 2; ++nt) {
        v2f a1 = {af[mt].z, af[mt].w};
        v2f b1 = {bf[nt].z, bf[nt].w};
        acc[mt][nt] = __builtin_amdgcn_wmma_f32_16x16x4_f32(
            false, a1, false, b1, (short)0, acc[mt][nt], false, false);
      }
  }

  // out = 1 - dot + lambda*mi  ==  (1 + lambda*mi) - dot
  const float c_add = 1.0f + LAMBDA_IB * mi_p[0];
  #pragma unroll
  for (int mt = 0; mt < 4; ++mt) {
    #pragma unroll
    for (int nt = 0; nt < 2; ++nt) {
      #pragma unroll
      for (int r = 0; r < 8; ++r) {
        // C/D layout: VGPR r, lanes 0-15 -> M=r, lanes 16-31 -> M=r+8; N = laneM
        const int orow = rowWave + mt * 16 + r + 8 * laneHi;
        const int ocol = colWave + nt * 16 + laneM;
        out[(size_t)orow * M_ROWS + ocol] = c_add - acc[mt][nt][r];
      }
    }
  }
}

extern "C" void kernel_launch(void* const* d_in, const int* in_sizes, int n_in,
                              void* d_out, int out_size, void* d_ws, size_t ws_size,
                              hipStream_t stream) {
  const float* x1 = (const float*)d_in[0];
  const float* x2 = (const float*)d_in[1];
  const float* Wc = (const float*)d_in[2];
  const float* bc = (const float*)d_in[3];
  const float* W1 = (const float*)d_in[4];
  const float* b1 = (const float*)d_in[5];
  const float* W2 = (const float*)d_in[6];
  const float* b2 = (const float*)d_in[7];
  const float* W3 = (const float*)d_in[8];
  const float* b3 = (const float*)d_in[9];
  const int* perm = (const int*)d_in[10];
  float* out = (float*)d_out;

  // Workspace layout (floats): ~25.3 MB total
  float* ws   = (float*)d_ws;
  float* x1sw = ws;                               // 8192*256 (swizzled fragments)
  float* x2sw = x1sw + (size_t)N_ROWS * D_DIM;    // 8192*256
  float* xc1  = x2sw + (size_t)M_ROWS * D_DIM;    // 8192*128
  float* xc2  = xc1 + (size_t)N_ROWS * C_DIM;     // 8192*128
  float* t_xy = xc2 + (size_t)M_ROWS * C_DIM;     // 8192
  float* t_sh = t_xy + N_ROWS;                    // 8192
  float* mi   = t_sh + N_ROWS;                    // 1

  rownorm_swizzle_kernel<<<N_ROWS, 256, 0, stream>>>(x1, x1sw);
  rownorm_swizzle_kernel<<<M_ROWS, 256, 0, stream>>>(x2, x2sw);
  compress_kernel<<<N_ROWS, 128, 0, stream>>>(x1, Wc, bc, xc1);
  compress_kernel<<<M_ROWS, 128, 0, stream>>>(x2, Wc, bc, xc2);
  critic_kernel<<<N_ROWS, 128, 0, stream>>>(xc1, xc2, W1, b1, W2, b2, W3, b3, perm, 0, t_xy);
  critic_kernel<<<N_ROWS, 128, 0, stream>>>(xc1, xc2, W1, b1, W2, b2, W3, b3, perm, 1, t_sh);
  mi_reduce_kernel<<<1, 256, 0, stream>>>(t_xy, t_sh, mi);

  dim3 grid(M_ROWS / 64, N_ROWS / 256);           // 128 x 32
  dist_wmma_kernel<<<grid, 256, 0, stream>>>(x1sw, x2sw, mi, out);
}